// dpLGAR_49924699849066
// MI455X (gfx1250) — compile-verified
//
#include <hip/hip_runtime.h>
#include <cstdint>
#include <cstddef>

// ---------------- LGAR constants (match reference) ----------------
#define DT_H        (1.0f / 12.0f)
#define INV_DEPTH   0.02f          // 1 / LAYER_DEPTH_CM (50)
#define PONDED_MAX  2.0f
#define PSI0        2000.0f
#define PSI_WILT    15495.0f

static constexpr int CPT = 2;   // catchments per thread (ILP in the serial chain)
static constexpr int BLK = 64;  // 2 waves per workgroup, spread across WGPs

// Native transcendental helpers (v_log_f32 returns log2, v_exp_f32 is 2^x)
__device__ __forceinline__ float fexp2(float x) { return __builtin_amdgcn_exp2f(x); }
__device__ __forceinline__ float flog2(float x) { return __builtin_amdgcn_logf(x);  }
__device__ __forceinline__ float fsqrt(float x) { return __builtin_amdgcn_sqrtf(x); }
__device__ __forceinline__ float fclampf(float v, float lo, float hi) {
  return fminf(fmaxf(v, lo), hi);
}

// van Genuchten theta(psi): tr + (te-tr) * (1 + (a*psi)^n)^(-m)
__device__ __forceinline__ float vg_theta_dev(float psi, float a, float n,
                                              float tr, float te) {
  float m   = 1.0f - 1.0f / n;
  float apn = fexp2(n * flog2(a * psi));          // (a*psi)^n
  float f   = fexp2(-m * flog2(1.0f + apn));      // (1+x)^(-m)
  return tr + (te - tr) * f;
}

__global__ __launch_bounds__(BLK) void lgar_scan_kernel(
    const float* __restrict__ x,        // [B,2] precip, pet (mm/h)
    const float* __restrict__ alpha,    // [3]
    const float* __restrict__ nvg,      // [3]
    const float* __restrict__ ksat,     // [3]
    const float* __restrict__ theta_r,  // [3]
    const float* __restrict__ theta_e,  // [3]
    const int*  __restrict__ p_nsteps,  // [1]
    const int*  __restrict__ p_nsub,    // [1]
    float* __restrict__ out,            // [B, nsteps]
    int B)
{
  const int tid = blockIdx.x * blockDim.x + threadIdx.x;
  const int b0  = tid * CPT;
  if (b0 >= B) return;

  const int nsteps = p_nsteps[0];
  const int nsub   = p_nsub[0];

  // ---- wave-uniform per-layer constants (SGPR-resident) ----
  float a[3], nn[3], tr[3], te[3], mm[3], invm[3], kdt[3], invdt[3], th_init[3];
#pragma unroll
  for (int l = 0; l < 3; ++l) {
    a[l]      = alpha[l];
    nn[l]     = nvg[l];
    tr[l]     = theta_r[l];
    te[l]     = theta_e[l];
    mm[l]     = 1.0f - 1.0f / nn[l];
    invm[l]   = 1.0f / mm[l];
    kdt[l]    = ksat[l] * DT_H;
    invdt[l]  = 1.0f / (te[l] - tr[l]);
    th_init[l] = vg_theta_dev(PSI0, a[l], nn[l], tr[l], te[l]);
  }
  const float thwp0   = vg_theta_dev(PSI_WILT, a[0], nn[0], tr[0], te[0]);
  const float inv_awp = 1.0f / (te[0] - thwp0);

  // ---- load forcings: 16B/lane, coalesced. Prefer CDNA5 async-to-LDS path ----
  __builtin_prefetch(x + 2 * b0, 0, 3);  // global_prefetch_b8

  float4 xf;
#if defined(__gfx1250__) && __has_builtin(__builtin_amdgcn_global_load_async_to_lds_b128)
  {
    __shared__ float sx[4 * BLK];
    typedef int v4i __attribute__((ext_vector_type(4)));
    typedef v4i __attribute__((address_space(1)))* gv4ip_t;  // global (AS1) v4i*
    typedef v4i __attribute__((address_space(3)))* lv4ip_t;  // LDS (AS3) v4i*
    gv4ip_t g = (gv4ip_t)(x + 2 * b0);
    lv4ip_t l = (lv4ip_t)(&sx[4 * threadIdx.x]);
    __builtin_amdgcn_global_load_async_to_lds_b128(g, l, 0, 0);
#if __has_builtin(__builtin_amdgcn_s_wait_asynccnt)
    __builtin_amdgcn_s_wait_asynccnt(0);
#else
    asm volatile("s_wait_asynccnt 0" ::: "memory");
#endif
    __syncthreads();
    xf = *reinterpret_cast<const float4*>(&sx[4 * threadIdx.x]);
  }
#else
  xf = *reinterpret_cast<const float4*>(x + 2 * b0);
#endif

  float precip[CPT], pet[CPT];
  precip[0] = xf.x * DT_H;  pet[0] = xf.y * DT_H;
  precip[1] = xf.z * DT_H;  pet[1] = xf.w * DT_H;

  // ---- register-resident scan state ----
  float th[CPT][3], pond[CPT];
#pragma unroll
  for (int c = 0; c < CPT; ++c) {
    pond[c] = 0.0f;
#pragma unroll
    for (int l = 0; l < 3; ++l) th[c][l] = th_init[l];
  }

  for (int t = 0; t < nsteps; ++t) {
    float acc[CPT];
#pragma unroll
    for (int c = 0; c < CPT; ++c) acc[c] = 0.0f;

    for (int j = 0; j < nsub; ++j) {
#pragma unroll
      for (int c = 0; c < CPT; ++c) {
        float pd    = precip[c] + pond[c];
        float avail = fclampf((th[c][0] - thwp0) * inv_awp, 0.0f, 1.0f);
        float aet   = pet[c] * avail;
        float run   = fmaxf(pd - PONDED_MAX, 0.0f);
        float water = pd - run;
        float cap   = kdt[0] * (1.0f + 10.0f * (te[0] - th[c][0]));
        float infil = fminf(water, cap);
        pond[c]     = water - infil;

        // Mualem-van Genuchten K(theta)*dt per layer (branch-free pow via exp2/log2)
        float perc[3];
#pragma unroll
        for (int l = 0; l < 3; ++l) {
          float se   = fclampf((th[c][l] - tr[l]) * invdt[l], 1e-6f, 1.0f);
          float l2se = flog2(se);
          float t1   = fexp2(l2se * invm[l]);          // se^(1/m)
          float om   = fmaxf(1.0f - t1, 0.0f);
          float t2   = fexp2(mm[l] * flog2(om));       // om^m ; log2(0)->-inf -> 0
          float fac  = 1.0f - t2;
          perc[l]    = kdt[l] * fsqrt(se) * (fac * fac);
        }

        th[c][0] = fclampf(th[c][0] + (infil - aet - perc[0]) * INV_DEPTH, tr[0], te[0]);
        th[c][1] = fclampf(th[c][1] + (perc[0] - perc[1]) * INV_DEPTH,     tr[1], te[1]);
        th[c][2] = fclampf(th[c][2] + (perc[1] - perc[2]) * INV_DEPTH,     tr[2], te[2]);

        acc[c] += run;
      }
    }

#pragma unroll
    for (int c = 0; c < CPT; ++c) {
      int b = b0 + c;
      if (b < B) {
        __builtin_nontemporal_store(acc[c], out + (size_t)b * nsteps + t);
      }
    }
  }
}

extern "C" void kernel_launch(void* const* d_in, const int* in_sizes, int n_in,
                              void* d_out, int out_size, void* d_ws, size_t ws_size,
                              hipStream_t stream) {
  const float* x       = (const float*)d_in[0];
  const float* alpha   = (const float*)d_in[1];
  const float* nvg     = (const float*)d_in[2];
  const float* ksat    = (const float*)d_in[3];
  const float* theta_r = (const float*)d_in[4];
  const float* theta_e = (const float*)d_in[5];
  const int*   nsteps  = (const int*)d_in[6];
  const int*   nsub    = (const int*)d_in[7];

  const int B       = in_sizes[0] / 2;             // x is [B,2]
  const int threads = (B + CPT - 1) / CPT;
  dim3 block(BLK);
  dim3 grid((threads + BLK - 1) / BLK);

  lgar_scan_kernel<<<grid, block, 0, stream>>>(
      x, alpha, nvg, ksat, theta_r, theta_e, nsteps, nsub,
      (float*)d_out, B);
}